// SimpleNet_2851858284831
// MI455X (gfx1250) — compile-verified
//
#include <hip/hip_runtime.h>
#include <hip/hip_bf16.h>

typedef float v2f __attribute__((ext_vector_type(2)));
typedef float v8f __attribute__((ext_vector_type(8)));

#define NEGV (-1e9f)

constexpr int BD = 256, HD = 64, WD = 64;
constexpr int HW = HD * WD;                 // 4096
constexpr int NPOS = BD * HW;               // 1048576
constexpr int OFF_LOGP = 0;
constexpr int OFF_CRITIC = 256;
constexpr int OFF_FAC = 512;
constexpr long long OFF_UNIT = OFF_FAC + (long long)NPOS;          // 1049088
constexpr long long OFF_ENT  = OFF_UNIT + (long long)NPOS * 6;     // 7340544

// Concatenated head rows in the 80-wide (padded from 73) logit vector:
// [0,4) Wf | [4,11) Wt | [11,16) Wdir0 | [16,21) Wdir1 | [21,26) Wres0 |
// [26,31) Wres1 | [31,41) Wamt0 | [41,51) Wamt1 | [51,61) Wamt2 |
// [61,73) Wrep0..5 (2 each) | [73,80) zero pad

template<int N>
__device__ inline void catN(const float* lg, unsigned mask, int act,
                            float& lp, float& ent) {
  float ml[N];
#pragma unroll
  for (int i = 0; i < N; i++) ml[i] = ((mask >> i) & 1u) ? lg[i] : NEGV;
  float mx = ml[0];
#pragma unroll
  for (int i = 1; i < N; i++) mx = fmaxf(mx, ml[i]);
  float s = 0.f;
#pragma unroll
  for (int i = 0; i < N; i++) s += __expf(ml[i] - mx);
  float lse = mx + __logf(s);
  float e = 0.f;
#pragma unroll
  for (int i = 0; i < N; i++) {
    if ((mask >> i) & 1u) { float l = ml[i] - lse; e -= __expf(l) * l; }
  }
  float sel = NEGV;
#pragma unroll
  for (int i = 0; i < N; i++) sel = (i == act) ? ml[i] : sel;
  ent = e;
  lp = sel - lse;
}

__global__ void init_kernel(const float* __restrict__ xc, const float* __restrict__ Wc,
                            const float* __restrict__ bc, float* __restrict__ out) {
  int i = threadIdx.x;  // 256 threads
  out[OFF_LOGP + i] = 0.f;
  out[OFF_ENT + i] = 0.f;
  float acc = bc[0];
#pragma unroll
  for (int k = 0; k < 4; k++) acc += xc[i * 4 + k] * Wc[k];
  out[OFF_CRITIC + i] = acc;
}

__global__ __launch_bounds__(128)
void actor_kernel(const float* __restrict__ xg,
                  const float* __restrict__ Wf, const float* __restrict__ bf,
                  const float* __restrict__ Wt, const float* __restrict__ bt,
                  const float* __restrict__ Wdir, const float* __restrict__ bdir,
                  const float* __restrict__ Wres, const float* __restrict__ bres,
                  const float* __restrict__ Wamt, const float* __restrict__ bamt,
                  const float* __restrict__ Wrep, const float* __restrict__ brep,
                  const unsigned char* __restrict__ vaf,  // [N,4]
                  const unsigned char* __restrict__ vam,  // [N,5,2]
                  const unsigned char* __restrict__ vat,  // [N,5,5,2]
                  const unsigned char* __restrict__ vap,  // [N,5,2]
                  const unsigned char* __restrict__ vad,  // [N,2]
                  const unsigned char* __restrict__ vas,  // [N,2]
                  const unsigned char* __restrict__ var_, // [N,2]
                  const unsigned char* __restrict__ van,  // [N]
                  const int* __restrict__ actf, const int* __restrict__ actt,
                  const int* __restrict__ actd, const int* __restrict__ actr,
                  const int* __restrict__ acta, const int* __restrict__ actp,
                  float* __restrict__ out) {
  __shared__ float sW[80 * 4];
  __shared__ float sB[80];
  __shared__ float sLog[128 * 83];  // stride 83 (odd) -> conflict-free phase-2 reads
  __shared__ float sRed0[128];
  __shared__ float sRed1[128];

  const int tid = threadIdx.x;
  const int blockBase = blockIdx.x * 128;

  // ---- stage head weights (80 padded rows x 4) + biases into LDS ----
  if (tid < 80) {
    const float* src = nullptr;
    float bias = 0.f;
    int j = tid;
    if (j < 4)       { src = Wf + j * 4;                 bias = bf[j]; }
    else if (j < 11) { int k = j - 4;  src = Wt + k * 4;          bias = bt[k]; }
    else if (j < 16) { int k = j - 11; src = Wdir + k * 4;        bias = bdir[k]; }
    else if (j < 21) { int k = j - 16; src = Wdir + (5 + k) * 4;  bias = bdir[5 + k]; }
    else if (j < 26) { int k = j - 21; src = Wres + k * 4;        bias = bres[k]; }
    else if (j < 31) { int k = j - 26; src = Wres + (5 + k) * 4;  bias = bres[5 + k]; }
    else if (j < 41) { int k = j - 31; src = Wamt + k * 4;        bias = bamt[k]; }
    else if (j < 51) { int k = j - 41; src = Wamt + (10 + k) * 4; bias = bamt[10 + k]; }
    else if (j < 61) { int k = j - 51; src = Wamt + (20 + k) * 4; bias = bamt[20 + k]; }
    else if (j < 73) { int k = j - 61; src = Wrep + k * 4;        bias = brep[k]; }
    if (src) {
      sW[j * 4 + 0] = src[0]; sW[j * 4 + 1] = src[1];
      sW[j * 4 + 2] = src[2]; sW[j * 4 + 3] = src[3];
    } else {
      sW[j * 4 + 0] = 0.f; sW[j * 4 + 1] = 0.f; sW[j * 4 + 2] = 0.f; sW[j * 4 + 3] = 0.f;
    }
    sB[j] = bias;
  }
  __syncthreads();

  // ---- phase 1: logits via V_WMMA_F32_16X16X4_F32 ----
  // A (16x4, MxK): lane<16 -> M=lane, K={0,1}; lane>=16 -> M=lane-16, K={2,3}
  // B (4x16, KxN): same K split across lane halves, N = lane&15
  // D (16x16):     VGPR r -> M = r + 8*(lane>>4), N = lane&15
  const int wave = tid >> 5, lane = tid & 31;
  const int m = lane & 15, kh = lane >> 4;
#pragma unroll
  for (int mt = 0; mt < 2; mt++) {
    const int posLocal = wave * 32 + mt * 16 + m;
    const long long pos = blockBase + posLocal;
    v2f a;
    a[0] = xg[pos * 4 + 2 * kh + 0];
    a[1] = xg[pos * 4 + 2 * kh + 1];
#pragma unroll
    for (int t = 0; t < 5; t++) {
      const int row = t * 16 + m;  // logit index this lane produces (N)
      v2f bm;
      bm[0] = sW[row * 4 + 2 * kh + 0];
      bm[1] = sW[row * 4 + 2 * kh + 1];
      v8f c = {0.f, 0.f, 0.f, 0.f, 0.f, 0.f, 0.f, 0.f};
      c = __builtin_amdgcn_wmma_f32_16x16x4_f32(false, a, false, bm,
                                                (short)0, c, false, false);
      const float bias = sB[row];
#pragma unroll
      for (int r = 0; r < 8; r++) {
        const int mrow = wave * 32 + mt * 16 + r + 8 * kh;  // position within block
        sLog[mrow * 83 + row] = c[r] + bias;
      }
    }
  }
  __syncthreads();

  // ---- phase 2: one thread per position, masked categorical math ----
  const long long g = blockBase + tid;
  const float* lg = &sLog[tid * 83];

  unsigned fmask = 0;
#pragma unroll
  for (int j = 0; j < 4; j++) fmask |= (vaf[g * 4 + j] ? 1u : 0u) << j;
  unsigned movem = 0;
#pragma unroll
  for (int j = 0; j < 10; j++) movem |= (vam[g * 10 + j] ? 1u : 0u) << j;
  unsigned long long trm = 0;
  for (int j = 0; j < 50; j++) trm |= (vat[g * 50 + j] ? 1ull : 0ull) << j;
  unsigned pkm = 0;
#pragma unroll
  for (int j = 0; j < 10; j++) pkm |= (vap[g * 10 + j] ? 1u : 0u) << j;
  unsigned digm = (vad[g * 2] ? 1u : 0u) | (vad[g * 2 + 1] ? 2u : 0u);
  unsigned sdm  = (vas[g * 2] ? 1u : 0u) | (vas[g * 2 + 1] ? 2u : 0u);
  unsigned rcm  = (var_[g * 2] ? 1u : 0u) | (var_[g * 2 + 1] ? 2u : 0u);
  unsigned dn   = van[g] ? 1u : 0u;

  const int af = actf[g], at = actt[g], ad = actd[g];
  const int ars = actr[g], aa = acta[g], arp = actp[g];

  unsigned tv = (movem ? 1u : 0) | (trm ? 2u : 0) | (pkm ? 4u : 0) |
                (digm ? 8u : 0) | (sdm ? 16u : 0) | (rcm ? 32u : 0) | (dn ? 64u : 0);
  unsigned d0m = 0, d1m = 0, s1m = 0, s2m = 0;
#pragma unroll
  for (int d = 0; d < 5; d++) {
    if ((movem >> (2 * d)) & 3u) d0m |= 1u << d;
    if ((trm >> (10 * d)) & 0x3FFull) d1m |= 1u << d;
  }
  const unsigned td = (unsigned)((trm >> (10 * ad)) & 0x3FFull);
#pragma unroll
  for (int s = 0; s < 5; s++) {
    if ((td >> (2 * s)) & 3u) s1m |= 1u << s;
    if ((pkm >> (2 * s)) & 3u) s2m |= 1u << s;
  }
  const unsigned r0m = (movem >> (2 * ad)) & 3u;
  const unsigned r1m = (td >> (2 * ars)) & 3u;
  const unsigned r2m = (pkm >> (2 * ars)) & 3u;

  float lpf, ef, lpt, et;
  catN<4>(lg + 0, fmask, af, lpf, ef);
  catN<7>(lg + 4, tv, at, lpt, et);
  float lpd0, ed0, lpr0, er0, lpd1, ed1, lps1, es1, lpa1, ea1, lpr1, er1;
  float lps2, es2, lpa2, ea2, lpr2, er2, lpr3, er3, lpr4, er4, lpa5, ea5, lpr5, er5;
  catN<5>(lg + 11, d0m, ad, lpd0, ed0);
  catN<2>(lg + 61, r0m, arp, lpr0, er0);
  catN<5>(lg + 16, d1m, ad, lpd1, ed1);
  catN<5>(lg + 21, s1m, ars, lps1, es1);
  catN<10>(lg + 31, 0x3FFu, aa, lpa1, ea1);
  catN<2>(lg + 63, r1m, arp, lpr1, er1);
  catN<5>(lg + 26, s2m, ars, lps2, es2);
  catN<10>(lg + 41, 0x3FFu, aa, lpa2, ea2);
  catN<2>(lg + 65, r2m, arp, lpr2, er2);
  catN<2>(lg + 67, digm, arp, lpr3, er3);
  catN<2>(lg + 69, sdm, arp, lpr4, er4);
  catN<10>(lg + 51, 0x3FFu, aa, lpa5, ea5);
  catN<2>(lg + 71, rcm, arp, lpr5, er5);

  float lpp = 0.f, epp = 0.f;
  switch (at) {
    case 0: lpp = lpd0 + lpr0;               epp = ed0 + er0;               break;
    case 1: lpp = lpd1 + lps1 + lpa1 + lpr1; epp = ed1 + es1 + ea1 + er1;   break;
    case 2: lpp = lps2 + lpa2 + lpr2;        epp = es2 + ea2 + er2;         break;
    case 3: lpp = lpr3;                      epp = er3;                     break;
    case 4: lpp = lpr4;                      epp = er4;                     break;
    case 5: lpp = lpa5 + lpr5;               epp = ea5 + er5;               break;
    default: break;
  }
  const bool fm = fmask != 0u, um = tv != 0u;
  const float lpc = (fm ? lpf : 0.f) + (um ? (lpt + lpp) : 0.f);
  const float enc = (fm ? ef : 0.f) + (um ? (et + epp) : 0.f);

  // ---- per-position outputs ----
  out[OFF_FAC + g] = fm ? (float)af : 0.f;
  const bool hd = (at == 0) || (at == 1);
  const bool hs = (at == 1) || (at == 2);
  const bool ha = (at == 1) || (at == 2) || (at == 5);
  const bool hr = (at <= 5);
  float2* up = (float2*)(out + OFF_UNIT + g * 6);  // 8B-aligned: OFF_UNIT and g*6 even
  up[0] = make_float2(um ? (float)at : 0.f, (um && hd) ? (float)ad : 0.f);
  up[1] = make_float2((um && hs) ? (float)ars : 0.f, (um && ha) ? (float)aa : 0.f);
  up[2] = make_float2((um && hr) ? (float)arp : 0.f, um ? 1.f : 0.f);

  // ---- block reduction -> per-batch atomics (block lies within one b) ----
  sRed0[tid] = lpc;
  sRed1[tid] = enc;
  __syncthreads();
  for (int off = 64; off > 0; off >>= 1) {
    if (tid < off) { sRed0[tid] += sRed0[tid + off]; sRed1[tid] += sRed1[tid + off]; }
    __syncthreads();
  }
  if (tid == 0) {
    const int b = blockBase / HW;
    atomicAdd(&out[OFF_LOGP + b], sRed0[0]);
    atomicAdd(&out[OFF_ENT + b], sRed1[0]);
  }
}

extern "C" void kernel_launch(void* const* d_in, const int* in_sizes, int n_in,
                              void* d_out, int out_size, void* d_ws, size_t ws_size,
                              hipStream_t stream) {
  const float* xg   = (const float*)d_in[0];
  const float* xc   = (const float*)d_in[1];
  const float* Wc   = (const float*)d_in[2];
  const float* bc   = (const float*)d_in[3];
  const float* Wf   = (const float*)d_in[4];
  const float* bf   = (const float*)d_in[5];
  const float* Wt   = (const float*)d_in[6];
  const float* bt   = (const float*)d_in[7];
  const float* Wdir = (const float*)d_in[8];
  const float* bdir = (const float*)d_in[9];
  const float* Wres = (const float*)d_in[10];
  const float* bres = (const float*)d_in[11];
  const float* Wamt = (const float*)d_in[12];
  const float* bamt = (const float*)d_in[13];
  const float* Wrep = (const float*)d_in[14];
  const float* brep = (const float*)d_in[15];
  const unsigned char* vaf  = (const unsigned char*)d_in[16];
  const unsigned char* vam  = (const unsigned char*)d_in[17];
  const unsigned char* vat  = (const unsigned char*)d_in[18];
  const unsigned char* vap  = (const unsigned char*)d_in[19];
  const unsigned char* vad  = (const unsigned char*)d_in[20];
  const unsigned char* vas  = (const unsigned char*)d_in[21];
  const unsigned char* var_ = (const unsigned char*)d_in[22];
  const unsigned char* van  = (const unsigned char*)d_in[23];
  const int* actf = (const int*)d_in[24];
  const int* actt = (const int*)d_in[25];
  const int* actd = (const int*)d_in[26];
  const int* actr = (const int*)d_in[27];
  const int* acta = (const int*)d_in[28];
  const int* actp = (const int*)d_in[29];
  float* out = (float*)d_out;

  hipLaunchKernelGGL(init_kernel, dim3(1), dim3(256), 0, stream, xc, Wc, bc, out);
  hipLaunchKernelGGL(actor_kernel, dim3(NPOS / 128), dim3(128), 0, stream,
                     xg, Wf, bf, Wt, bt, Wdir, bdir, Wres, bres, Wamt, bamt,
                     Wrep, brep, vaf, vam, vat, vap, vad, vas, var_, van,
                     actf, actt, actd, actr, acta, actp, out);
}